// OverAll_48816598286825
// MI455X (gfx1250) — compile-verified
//
#include <hip/hip_runtime.h>
#include <math.h>

#define DFEAT 100
#define EPSF  1e-12f

typedef float v2f __attribute__((ext_vector_type(2)));
typedef float v4f __attribute__((ext_vector_type(4)));
typedef float v8f __attribute__((ext_vector_type(8)));

// ---------------- normalize none_relation ----------------
__global__ void k_norm_nr(const float* __restrict__ nrel, float* __restrict__ nr) {
    __shared__ float red[128];
    int t = threadIdx.x;
    float x = (t < DFEAT) ? nrel[t] : 0.f;
    red[t] = x * x;
    __syncthreads();
    for (int s = 64; s > 0; s >>= 1) {
        if (t < s) red[t] += red[t + s];
        __syncthreads();
    }
    float inv = 1.f / fmaxf(sqrtf(red[0]), EPSF);
    if (t < DFEAT) nr[t] = x * inv;
}

// ---------------- depth-0 slice: tanh(features) ----------------
__global__ void k_tanh(const float* __restrict__ f, float* __restrict__ out,
                       long long tot, int pitch) {
    long long i = (long long)blockIdx.x * 256 + threadIdx.x;
    if (i >= tot) return;
    long long n = i / DFEAT;
    int k = (int)(i - n * DFEAT);
    out[n * pitch + k] = tanhf(f[i]);
}

// ---------------- CSR row pointers (rows are sorted) ----------------
__global__ void k_rowptr(const long long* __restrict__ adj, int* __restrict__ row_ptr,
                         int N, int E) {
    int r = blockIdx.x * 256 + threadIdx.x;
    if (r > N) return;
    int lo = 0, hi = E;
    while (lo < hi) {
        int mid = (lo + hi) >> 1;
        if (adj[2 * (long long)mid] < (long long)r) lo = mid + 1; else hi = mid;
    }
    row_ptr[r] = lo;
}

// ---------------- edge scores via WMMA (16 edges / wave) ----------------
// A (16x4) row 0 = nr chunk: lanes 0/16 point at the real nr in LDS, all other
// lanes point at a zero LDS array -> unconditional ds loads, no exec masking.
// B (4x16) = gathered edge features: lane l and l+16 share edge (l&15); a b64
// load at frow + kb + khalf lands each lane's two B elements directly in the
// hardware layout (partner lanes fetch adjacent 8B of the same 16B segment).
// D[0][n] accumulates the 16 dots; read back as c[0] on lanes 0..15.
__global__ void k_score(const float* __restrict__ base, int pitch,
                        const long long* __restrict__ adj,
                        const float* __restrict__ nr,
                        float* __restrict__ scores, int E) {
    __shared__ float s_nr[2 * DFEAT];          // [0..99]=0, [100..199]=nr
    if (threadIdx.x < DFEAT) {
        s_nr[threadIdx.x] = 0.f;
        s_nr[DFEAT + threadIdx.x] = nr[threadIdx.x];
    }
    __syncthreads();

    int lane = threadIdx.x & 31;
    int gw   = blockIdx.x * 8 + (threadIdx.x >> 5);
    int ebase = gw * 16;
    int m = lane & 15;                 // edge slot (B-matrix column N)
    int e = ebase + m;
    bool valid = (e < E);
    int ec = valid ? e : (E - 1);      // clamp: load real data, never store it
    int col = (int)adj[2 * (long long)ec + 1];
    const float* frow = base + (long long)col * pitch;
    int khalf = (lane >> 4) << 1;      // 0 for lanes 0-15, 2 for lanes 16-31
    // lanes 0/16 read real nr (+khalf), everyone else reads the zero array
    const float* nrp = s_nr + ((m == 0) ? DFEAT : 0) + khalf;

    v8f c = {};                        // D row 0 (lanes 0..15 of c[0]) = dots
    float sumsq = 0.f;
#pragma unroll
    for (int kb = 0; kb < DFEAT; kb += 4) {
        v2f b;                         // B[k][n]: lane n, K = khalf(+1)
        b.x = frow[kb + khalf];
        b.y = frow[kb + khalf + 1];
        sumsq += b.x * b.x + b.y * b.y;
        v2f a;                         // A[0][k] = nr[kb+k], rows 1..15 = 0
        a.x = nrp[kb];
        a.y = nrp[kb + 1];
        c = __builtin_amdgcn_wmma_f32_16x16x4_f32(false, a, false, b,
                                                  (short)0, c, false, false);
    }
    // per-edge sum of squares: combine lane m with lane m+16
    sumsq += __shfl_xor(sumsq, 16, 32);

    float dot = c[0];                  // D[0][N=lane] for lanes 0..15
    if (lane < 16 && valid) {
        float nrm = fmaxf(sqrtf(sumsq), EPSF);
        scores[e] = -dot / nrm;
    }
}

// ---------------- per-row softmax + weighted aggregation (1 wave / row) ----
__global__ void k_agg(const float* __restrict__ src, float* __restrict__ dst,
                      int pitch,
                      const long long* __restrict__ adj,
                      const int* __restrict__ row_ptr,
                      const float* __restrict__ scores, int N) {
    int lane = threadIdx.x & 31;
    int wave = threadIdx.x >> 5;
    int r = blockIdx.x * 8 + wave;
    if (r >= N) return;
    int start = row_ptr[r], end = row_ptr[r + 1];
    float* orow = dst + (long long)r * pitch;
    bool act = (lane < 25);                 // 25 float4 lanes cover 100 dims
    int dl = act ? (lane * 4) : 96;         // clamped load offset for idle lanes
    if (start >= end) {                     // empty segment -> tanh(0) = 0
        if (act) { v4f z = {0.f, 0.f, 0.f, 0.f}; *(v4f*)(orow + dl) = z; }
        return;
    }
    // segment max
    float mx = -3.402823466e38f;
    for (int e = start + lane; e < end; e += 32) mx = fmaxf(mx, scores[e]);
    for (int off = 16; off > 0; off >>= 1) mx = fmaxf(mx, __shfl_xor(mx, off, 32));
    // segment denom
    float sum = 0.f;
    for (int e = start + lane; e < end; e += 32) sum += __expf(scores[e] - mx);
    for (int off = 16; off > 0; off >>= 1) sum += __shfl_xor(sum, off, 32);
    float inv = 1.f / sum;
    // weighted feature sum: one b128 per lane per edge
    v4f acc = {0.f, 0.f, 0.f, 0.f};
    for (int e = start; e < end; ++e) {
        int col = (int)adj[2 * (long long)e + 1];
        const float* frow = src + (long long)col * pitch;
        if (e + 1 < end) {
            int cn = (int)adj[2 * (long long)(e + 1) + 1];
            __builtin_prefetch(src + (long long)cn * pitch + dl, 0, 0);
        }
        v4f q = *(const v4f*)(frow + dl);
        float w = __expf(scores[e] - mx) * inv;
        acc.x += w * q.x;
        acc.y += w * q.y;
        acc.z += w * q.z;
        acc.w += w * q.w;
    }
    if (act) {
        v4f t;
        t.x = tanhf(acc.x);
        t.y = tanhf(acc.y);
        t.z = tanhf(acc.z);
        t.w = tanhf(acc.w);
        *(v4f*)(orow + dl) = t;
    }
}

extern "C" void kernel_launch(void* const* d_in, const int* in_sizes, int n_in,
                              void* d_out, int out_size, void* d_ws, size_t ws_size,
                              hipStream_t stream) {
    const float*     features = (const float*)d_in[0];
    /* d_in[1] = rel_emb: unused by the reference */
    const long long* adj      = (const long long*)d_in[2];
    const float*     nrel     = (const float*)d_in[3];
    float*           out      = (float*)d_out;

    const int D = DFEAT;
    const int N = in_sizes[0] / D;          // 50000
    const int E = in_sizes[2] / 2;          // 800000
    const int pitch = 3 * D;                // 300 (concat of 3 depth slices)

    // workspace carve: scores | row_ptr | nr
    char* ws = (char*)d_ws;
    float* scores = (float*)ws;
    size_t off = ((size_t)E * 4 + 255) & ~(size_t)255;
    int* row_ptr = (int*)(ws + off);
    off += (((size_t)(N + 1) * 4 + 255) & ~(size_t)255);
    float* nr = (float*)(ws + off);

    k_norm_nr<<<1, 128, 0, stream>>>(nrel, nr);

    long long tot = (long long)N * D;
    k_tanh<<<(int)((tot + 255) / 256), 256, 0, stream>>>(features, out, tot, pitch);

    k_rowptr<<<(N + 1 + 255) / 256, 256, 0, stream>>>(adj, row_ptr, N, E);

    int waves   = (E + 15) / 16;
    int sblocks = (waves + 7) / 8;          // 8 waves (256 thr) per block
    int ablocks = (N + 7) / 8;
    for (int d = 0; d < 2; ++d) {
        const float* srcb = out + d * D;
        float*       dstb = out + (d + 1) * D;
        k_score<<<sblocks, 256, 0, stream>>>(srcb, pitch, adj, nr, scores, E);
        k_agg<<<ablocks, 256, 0, stream>>>(srcb, dstb, pitch, adj, row_ptr, scores, N);
    }
}